// StructuredMAPLoss_49692771614790
// MI455X (gfx1250) — compile-verified
//
#include <hip/hip_runtime.h>
#include <stdint.h>

// StructuredMAPLoss for MI455X (gfx1250, wave32).
// B=1024 samples, C=128 classes. One workgroup per class; x column staged into
// LDS via CDNA5 async global->LDS loads; O(B^2) pairwise rank counting runs
// out of LDS with vectorized b128 broadcast reads. Deterministic.

#define Bn 1024
#define Cn 128
#define TPB 256
#define RPT 4   // rows per thread (TPB*RPT == Bn)

static __device__ __forceinline__ float block_reduce(float v, float* rbuf, int t) {
    rbuf[t] = v;
    __syncthreads();
    for (int s = TPB / 2; s > 0; s >>= 1) {
        if (t < s) rbuf[t] += rbuf[t + s];
        __syncthreads();
    }
    float r = rbuf[0];
    __syncthreads();
    return r;
}

__global__ __launch_bounds__(TPB)
void map_loss_class_kernel(const float* __restrict__ x,
                           const int* __restrict__ target,
                           const unsigned char* __restrict__ mask,
                           float* __restrict__ out,       // [1 + B*C]
                           float* __restrict__ loss_ws)   // [C]
{
    __shared__ __align__(16) float xs[Bn];
    __shared__ int   fl[Bn];                 // bit0 valid, bit1 pos, bit2 neg
    __shared__ __align__(16) int pri[Bn];    // (r << 3) | flags
    __shared__ unsigned long long scanbuf[TPB];
    __shared__ float rbuf[TPB];

    const int c = blockIdx.x;
    const int t = threadIdx.x;

    // ---- Phase 0: stage column c. x goes global->LDS via async DMA path ----
    for (int k = 0; k < RPT; ++k) {
        int i = t + k * TPB;
        const float* gp = x + (size_t)i * Cn + c;
        unsigned ldsOff = (unsigned)(uintptr_t)(&xs[i]);   // low 32 bits = LDS offset
        asm volatile("global_load_async_to_lds_b32 %0, %1, off"
                     :: "v"(ldsOff), "v"(gp)
                     : "memory");
        int tg = target[(size_t)i * Cn + c];
        int m  = mask[(size_t)i * Cn + c] ? 1 : 0;
        int pos = (m && tg == 1) ? 1 : 0;
        int neg = (m && tg == 0) ? 1 : 0;
        fl[i] = m | (pos << 1) | (neg << 2);
    }
    asm volatile("s_wait_asynccnt 0" ::: "memory");
    __syncthreads();

    // ---- Phase 1: packed inclusive scans of (valid, neg, pos) in 21-bit fields
    const unsigned long long M21 = (1ull << 21) - 1ull;
    unsigned long long loc[RPT];
    {
        unsigned long long run = 0ull;
        for (int k = 0; k < RPT; ++k) {
            int i = t * RPT + k;
            int f = fl[i];
            unsigned long long e = (unsigned long long)(f & 1)
                                 | ((unsigned long long)((f >> 2) & 1) << 21)   // neg
                                 | ((unsigned long long)((f >> 1) & 1) << 42);  // pos
            run += e;
            loc[k] = run;
        }
        scanbuf[t] = run;
    }
    __syncthreads();
    for (int off = 1; off < TPB; off <<= 1) {
        unsigned long long mine = scanbuf[t];
        unsigned long long add  = (t >= off) ? scanbuf[t - off] : 0ull;
        __syncthreads();
        scanbuf[t] = mine + add;
        __syncthreads();
    }
    unsigned long long excl = (t > 0) ? scanbuf[t - 1] : 0ull;
    unsigned long long tot  = scanbuf[TPB - 1];
    const int V = (int)(tot & M21);
    const int N = (int)((tot >> 21) & M21);
    const int P = (int)((tot >> 42) & M21);
    const bool active = (P > 0) && (N > 0);

    // r_i closed form: base = V - 2*csv + valid_i
    //   pos: base + 2*csn ; neg: base + 2*csp - 2P ; else: base
    for (int k = 0; k < RPT; ++k) {
        int i = t * RPT + k;
        unsigned long long cs = excl + loc[k];
        int csv = (int)(cs & M21);
        int csn = (int)((cs >> 21) & M21);
        int csp = (int)((cs >> 42) & M21);
        int f = fl[i];
        int base = V - 2 * csv + (f & 1);
        int r;
        if (f & 2)      r = base + 2 * csn;
        else if (f & 4) r = base + 2 * csp - 2 * P;
        else            r = base;
        pri[i] = (r << 3) | (f & 7);
        out[1 + (size_t)i * Cn + c] = (active && (f & 1)) ? (float)r : 0.0f;
    }
    __syncthreads();

    // ---- Phase 2: pairwise counting out of LDS (vectorized b128 broadcasts) --
    float rowX[RPT]; int rowR[RPT]; int rowF[RPT];
    int cValid[RPT], cPosX[RPT], cNegLt[RPT], cPosGt[RPT];
    for (int k = 0; k < RPT; ++k) {
        int i = t * RPT + k;
        int p = pri[i];
        rowX[k] = xs[i]; rowR[k] = p >> 3; rowF[k] = p & 7;
        cValid[k] = 0; cPosX[k] = 0; cNegLt[k] = 0; cPosGt[k] = 0;
    }
    const float4* xs4 = (const float4*)xs;
    const int4*   pr4 = (const int4*)pri;
    for (int jq = 0; jq < Bn / 4; ++jq) {
        float4 xv = xs4[jq];
        int4   pv = pr4[jq];
        float xj[4] = {xv.x, xv.y, xv.z, xv.w};
        int   pj[4] = {pv.x, pv.y, pv.z, pv.w};
        #pragma unroll
        for (int u = 0; u < 4; ++u) {
            int j  = jq * 4 + u;
            int fj = pj[u] & 7;
            int rj = pj[u] >> 3;
            int vj = fj & 1, po = (fj >> 1) & 1, ne = (fj >> 2) & 1;
            #pragma unroll
            for (int k = 0; k < RPT; ++k) {
                int i = t * RPT + k;
                // descending-x rank with stable tie-break (matches stable argsort)
                bool gtx = (xj[u] > rowX[k]) || ((xj[u] == rowX[k]) && (j < i));
                cValid[k] += (vj && gtx) ? 1 : 0;
                cPosX[k]  += (po && gtx) ? 1 : 0;
                cNegLt[k] += (ne && (rj < rowR[k])) ? 1 : 0;
                cPosGt[k] += (po && (rj > rowR[k])) ? 1 : 0;
            }
        }
    }

    const float Pf = (float)P, Nf = (float)N;
    float apSum = 0.f, t1 = 0.f, t2 = 0.f, sumPx = 0.f, sumNx = 0.f;
    for (int k = 0; k < RPT; ++k) {
        if (rowF[k] & 2) {              // positive row
            apSum += (float)(cPosX[k] + 1) / (float)(cValid[k] + 1);
            t1    += rowX[k] * (2.0f * (float)cNegLt[k] - Nf);
            sumPx += rowX[k];
        } else if (rowF[k] & 4) {       // negative row
            t2    += rowX[k] * (2.0f * (float)cPosGt[k] - Pf);
            sumNx += rowX[k];
        }
    }

    apSum = block_reduce(apSum, rbuf, t);
    t1    = block_reduce(t1,    rbuf, t);
    t2    = block_reduce(t2,    rbuf, t);
    sumPx = block_reduce(sumPx, rbuf, t);
    sumNx = block_reduce(sumNx, rbuf, t);

    if (t == 0) {
        float ap    = apSum / (Pf + 1e-8f);
        float denom = Pf * Nf + 1e-8f;
        float sgt   = (Nf * sumPx - Pf * sumNx) / denom;
        float slai  = (t1 - t2) / denom;
        loss_ws[c]  = active ? (1.0f - ap + slai - sgt) : 0.0f;
    }
}

__global__ void map_loss_finalize_kernel(const float* __restrict__ loss_ws,
                                         float* __restrict__ out)
{
    // single thread, fixed order -> deterministic
    float s = 0.0f;
    for (int c = 0; c < Cn; ++c) s += loss_ws[c];
    out[0] = s * (1.0f / (float)Cn);
}

extern "C" void kernel_launch(void* const* d_in, const int* in_sizes, int n_in,
                              void* d_out, int out_size, void* d_ws, size_t ws_size,
                              hipStream_t stream) {
    (void)in_sizes; (void)n_in; (void)out_size; (void)ws_size;
    const float*         x      = (const float*)d_in[0];
    const int*           target = (const int*)d_in[1];
    const unsigned char* mask   = (const unsigned char*)d_in[2];  // jnp bool_ = 1 byte
    float* out = (float*)d_out;       // [0] = loss, [1..] = ranking [B,C]
    float* lws = (float*)d_ws;        // C floats of scratch

    map_loss_class_kernel<<<Cn, TPB, 0, stream>>>(x, target, mask, out, lws);
    map_loss_finalize_kernel<<<1, 1, 0, stream>>>(lws, out);
}